// GATLayer_81037442940968
// MI455X (gfx1250) — compile-verified
//
#include <hip/hip_runtime.h>
#include <hip/hip_bf16.h>
#include <math.h>

// ---------------------------------------------------------------------------
// GAT (2 layers) for gfx1250.  GEMMs use V_WMMA_F32_16X16X4_F32 (full fp32
// matrix pipe).  Edge softmax+scatter is the memory-bound part: 3 passes with
// coalesced per-feature atomics (64 consecutive lanes -> consecutive floats).
// ---------------------------------------------------------------------------

typedef __attribute__((ext_vector_type(2))) float v2f;
typedef __attribute__((ext_vector_type(8))) float v8f;

#define FH 64            // hidden / output feature width (both layers)
#define NEG_SLOPE 0.2f

// ---------------- GEMM: [M x K] @ [K x 64] -> [M x 64], fp32 WMMA ----------
// One wave computes a 16-row strip across all 64 output columns (4 tiles of
// 16).  A-fragments (16x4) are loaded once per k-step and reused for all 4
// column tiles; W (<=64KB) lives in L2.
__global__ void gat_gemm_wmma_f32(const float* __restrict__ A,
                                  const float* __restrict__ B,
                                  float* __restrict__ C,
                                  int M, int K) {
    const int wave = (blockIdx.x * blockDim.x + threadIdx.x) >> 5;
    const int lane = threadIdx.x & 31;
    if (wave * 16 >= M) return;                 // uniform per wave

    const int half = lane >> 4;                 // 0: K+0..1, 1: K+2..3
    const int l    = lane & 15;                 // row (A) / col (B,C)

    const float* arow = A + (size_t)(wave * 16 + l) * K;

    v8f acc0 = {}, acc1 = {}, acc2 = {}, acc3 = {};

    for (int k = 0; k < K; k += 4) {
        v2f a;
        a.x = arow[k + 2 * half + 0];
        a.y = arow[k + 2 * half + 1];

        const float* b0 = B + (size_t)(k + 2 * half) * FH;   // row k+2h
        const float* b1 = b0 + FH;                           // row k+2h+1

        v2f bb;
        bb.x = b0[l];       bb.y = b1[l];
        acc0 = __builtin_amdgcn_wmma_f32_16x16x4_f32(false, a, false, bb,
                                                     (short)0, acc0, false, false);
        bb.x = b0[16 + l];  bb.y = b1[16 + l];
        acc1 = __builtin_amdgcn_wmma_f32_16x16x4_f32(false, a, false, bb,
                                                     (short)0, acc1, false, false);
        bb.x = b0[32 + l];  bb.y = b1[32 + l];
        acc2 = __builtin_amdgcn_wmma_f32_16x16x4_f32(false, a, false, bb,
                                                     (short)0, acc2, false, false);
        bb.x = b0[48 + l];  bb.y = b1[48 + l];
        acc3 = __builtin_amdgcn_wmma_f32_16x16x4_f32(false, a, false, bb,
                                                     (short)0, acc3, false, false);
    }

    // C/D layout: VGPR v holds row (v + 8*half), col = l (+ tile*16)
    float* crow = C + (size_t)(wave * 16) * FH;
    #pragma unroll
    for (int v = 0; v < 8; ++v) {
        const int m = v + half * 8;
        crow[(size_t)m * FH +  0 + l] = acc0[v];
        crow[(size_t)m * FH + 16 + l] = acc1[v];
        crow[(size_t)m * FH + 32 + l] = acc2[v];
        crow[(size_t)m * FH + 48 + l] = acc3[v];
    }
}

// ---------------- per-node attention logits: alpha_s, alpha_d --------------
// One wave per node: lane reads features (lane, lane+32), shfl reduction.
__global__ void gat_node_alpha(const float* __restrict__ xw,
                               const float* __restrict__ a_src,
                               const float* __restrict__ a_dst,
                               float* __restrict__ out_s,
                               float* __restrict__ out_d, int n) {
    const int wave = (blockIdx.x * blockDim.x + threadIdx.x) >> 5;
    const int lane = threadIdx.x & 31;
    if (wave >= n) return;
    const float* row = xw + (size_t)wave * FH;
    float r0 = row[lane], r1 = row[lane + 32];
    float s = r0 * a_src[lane] + r1 * a_src[lane + 32];
    float d = r0 * a_dst[lane] + r1 * a_dst[lane + 32];
    #pragma unroll
    for (int off = 16; off > 0; off >>= 1) {
        s += __shfl_down(s, off, 32);
        d += __shfl_down(d, off, 32);
    }
    if (lane == 0) { out_s[wave] = s; out_d[wave] = d; }
}

// ---------------- init: m = -inf, denom = 0, acc = 0 -----------------------
__global__ void gat_seg_init(float* __restrict__ m, float* __restrict__ denom,
                             float* __restrict__ acc, int n) {
    const int i = blockIdx.x * blockDim.x + threadIdx.x;
    if (i < n) { m[i] = -__builtin_inff(); denom[i] = 0.f; }
    if (i < n * FH) acc[i] = 0.f;
}

// atomic float max via signed/unsigned int ordering trick (init = -inf).
__device__ __forceinline__ void atomicMaxF(float* addr, float v) {
    if (v >= 0.f) atomicMax((int*)addr, __float_as_int(v));
    else          atomicMin((unsigned int*)addr, __float_as_uint(v));
}

__device__ __forceinline__ void edge_sd(const int* __restrict__ src,
                                        const int* __restrict__ dst,
                                        int i, int E, int& s, int& d) {
    if (i < E) { s = src[i]; d = dst[i]; }
    else       { s = d = i - E; }            // self loops appended
}

// ---------------- pass 1: segment max of leaky_relu logits -----------------
__global__ void gat_edge_max(const int* __restrict__ src, const int* __restrict__ dst,
                             const float* __restrict__ as, const float* __restrict__ ad,
                             float* __restrict__ m, int E, int n) {
    const int i = blockIdx.x * blockDim.x + threadIdx.x;
    if (i >= E + n) return;
    int s, d; edge_sd(src, dst, i, E, s, d);
    float v = as[s] + ad[d];
    v = v > 0.f ? v : NEG_SLOPE * v;
    atomicMaxF(&m[d], v);
}

// ---------------- pass 2: ex = exp(e - m[dst]); denom += ex ----------------
__global__ void gat_edge_sumexp(const int* __restrict__ src, const int* __restrict__ dst,
                                const float* __restrict__ as, const float* __restrict__ ad,
                                const float* __restrict__ m, float* __restrict__ denom,
                                float* __restrict__ ex, int E, int n) {
    const int i = blockIdx.x * blockDim.x + threadIdx.x;
    if (i >= E + n) return;
    int s, d; edge_sd(src, dst, i, E, s, d);
    float v = as[s] + ad[d];
    v = v > 0.f ? v : NEG_SLOPE * v;
    float e = __expf(v - m[d]);
    ex[i] = e;
    atomicAdd(&denom[d], e);
}

// ---------------- pass 3: acc[dst,f] += (ex/denom[dst]) * xw[src,f] --------
// 64 consecutive threads = one edge; consecutive lanes hit consecutive floats
// (coalesced gather + coalesced atomics).
__global__ void gat_edge_scatter(const int* __restrict__ src, const int* __restrict__ dst,
                                 const float* __restrict__ ex, const float* __restrict__ denom,
                                 const float* __restrict__ xw, float* __restrict__ acc,
                                 int E, int n) {
    const int tid = blockIdx.x * blockDim.x + threadIdx.x;
    const int f = tid & (FH - 1);
    const int i = tid >> 6;
    if (i >= E + n) return;
    int s, d; edge_sd(src, dst, i, E, s, d);
    const float coeff = ex[i] / denom[d];
    atomicAdd(&acc[(size_t)d * FH + f], coeff * xw[(size_t)s * FH + f]);
}

// ---------------- bias (+ optional ReLU), in place -------------------------
__global__ void gat_bias_act(float* __restrict__ y, const float* __restrict__ b,
                             int n, int do_relu) {
    const int i = blockIdx.x * blockDim.x + threadIdx.x;
    if (i >= n * FH) return;
    float v = y[i] + b[i & (FH - 1)];
    if (do_relu) v = v > 0.f ? v : 0.f;
    y[i] = v;
}

// ---------------------------------------------------------------------------
extern "C" void kernel_launch(void* const* d_in, const int* in_sizes, int n_in,
                              void* d_out, int out_size, void* d_ws, size_t ws_size,
                              hipStream_t stream) {
    const float* x    = (const float*)d_in[0];
    const int*   ei   = (const int*)  d_in[1];
    const float* W1   = (const float*)d_in[2];
    const float* a1s  = (const float*)d_in[3];
    const float* a1d  = (const float*)d_in[4];
    const float* b1   = (const float*)d_in[5];
    const float* W2   = (const float*)d_in[6];
    const float* a2s  = (const float*)d_in[7];
    const float* a2d  = (const float*)d_in[8];
    const float* b2   = (const float*)d_in[9];

    const int FIN = 256;
    const int n   = in_sizes[0] / FIN;      // 100000
    const int E   = in_sizes[1] / 2;        // 1600000
    const int Ep  = E + n;                  // with self loops
    const int* src = ei;
    const int* dst = ei + E;

    // workspace carve-out (floats): xw | h1 | as | ad | m | denom | ex
    float* ws   = (float*)d_ws;
    float* xw    = ws;  ws += (size_t)n * FH;   // layer GEMM output (reused)
    float* h1    = ws;  ws += (size_t)n * FH;   // layer-1 result / layer-2 input
    float* as    = ws;  ws += n;
    float* ad    = ws;  ws += n;
    float* m     = ws;  ws += n;
    float* denom = ws;  ws += n;
    float* ex    = ws;  ws += Ep;

    float* out = (float*)d_out;

    const int BLK = 256;
    const int wavesM   = (n + 15) / 16;                       // GEMM waves
    const int gGemm    = (wavesM + (BLK / 32) - 1) / (BLK / 32);
    const int gAlpha   = ((size_t)n * 32 + BLK - 1) / BLK;
    const int gInit    = (n * FH + BLK - 1) / BLK;
    const int gEdge    = (Ep + BLK - 1) / BLK;
    const int gScatter = (int)(((size_t)Ep * FH + BLK - 1) / BLK);
    const int gNF      = (n * FH + BLK - 1) / BLK;

    // ---------------- layer 1 ----------------
    gat_gemm_wmma_f32 <<<gGemm, BLK, 0, stream>>>(x, W1, xw, n, FIN);
    gat_node_alpha    <<<gAlpha, BLK, 0, stream>>>(xw, a1s, a1d, as, ad, n);
    gat_seg_init      <<<gInit, BLK, 0, stream>>>(m, denom, h1, n);
    gat_edge_max      <<<gEdge, BLK, 0, stream>>>(src, dst, as, ad, m, E, n);
    gat_edge_sumexp   <<<gEdge, BLK, 0, stream>>>(src, dst, as, ad, m, denom, ex, E, n);
    gat_edge_scatter  <<<gScatter, BLK, 0, stream>>>(src, dst, ex, denom, xw, h1, E, n);
    gat_bias_act      <<<gNF, BLK, 0, stream>>>(h1, b1, n, 0);

    // ---------------- layer 2 ----------------
    gat_gemm_wmma_f32 <<<gGemm, BLK, 0, stream>>>(h1, W2, xw, n, FH);
    gat_node_alpha    <<<gAlpha, BLK, 0, stream>>>(xw, a2s, a2d, as, ad, n);
    gat_seg_init      <<<gInit, BLK, 0, stream>>>(m, denom, out, n);
    gat_edge_max      <<<gEdge, BLK, 0, stream>>>(src, dst, as, ad, m, E, n);
    gat_edge_sumexp   <<<gEdge, BLK, 0, stream>>>(src, dst, as, ad, m, denom, ex, E, n);
    gat_edge_scatter  <<<gScatter, BLK, 0, stream>>>(src, dst, ex, denom, xw, out, E, n);
    gat_bias_act      <<<gNF, BLK, 0, stream>>>(out, b2, n, 1);
}